// DualGraphEncoder_75033078661165
// MI455X (gfx1250) — compile-verified
//
#include <hip/hip_runtime.h>
#include <math.h>

// CDNA5 / gfx1250 — wave32, WMMA fp32 path.
typedef __attribute__((ext_vector_type(2))) float v2f;
typedef __attribute__((ext_vector_type(8))) float v8f;

#define DOUT 256

// ---------------------------------------------------------------------------
// Edge-count: cnt[row[e]] += 1 via native global_atomic_add_f32
// ---------------------------------------------------------------------------
__global__ __launch_bounds__(256)
void count_edges_kernel(const int* __restrict__ row, float* __restrict__ cnt, int E)
{
    int e = blockIdx.x * blockDim.x + threadIdx.x;
    if (e < E) unsafeAtomicAdd(&cnt[row[e]], 1.0f);
}

// ---------------------------------------------------------------------------
// Scatter-add of neighbor features: agg[row[e], :] += X[col[e], :]
// One thread per (edge, 4-feature chunk): coalesced float4 gather + 4 fp32
// global atomics (agg is 51MB -> L2-resident on MI455X's 192MB L2).
// ---------------------------------------------------------------------------
__global__ __launch_bounds__(256)
void scatter_add_kernel(const float* __restrict__ X, const int* __restrict__ row,
                        const int* __restrict__ col, float* __restrict__ agg,
                        int E, int D)
{
    int chunks = D >> 2;
    long long t = (long long)blockIdx.x * blockDim.x + threadIdx.x;
    int e = (int)(t / chunks);
    int c = (int)(t - (long long)e * chunks);
    if (e >= E) return;
    int s = col[e], d = row[e];
    const float4 v = ((const float4*)(X + (long long)s * D))[c];
    float* p = agg + (long long)d * D + (c << 2);
    unsafeAtomicAdd(p + 0, v.x);
    unsafeAtomicAdd(p + 1, v.y);
    unsafeAtomicAdd(p + 2, v.z);
    unsafeAtomicAdd(p + 3, v.w);
}

// ---------------------------------------------------------------------------
// Fused SAGE layer: H = relu(X@Ws + (agg/cnt)@Wn + bs + bn)
// One wave per 16x16 output tile; K-loop of V_WMMA_F32_16X16X4_F32.
// ISA §7.12.2 layouts (wave32):
//   A (16x4 f32): lane L, vgpr v -> A[L%16][k + v + 2*(L/16)]   (float2/lane)
//   B (4x16 f32): lane L, vgpr v -> B[k + v + 2*(L/16)][L%16]
//   C/D (16x16) : lane L, vgpr v -> D[v + 8*(L/16)][L%16]
// mode: 0 -> H = relu(...) ; 1 -> H = sigmoid(alpha)*relu(...) (overwrite)
//       2 -> H += (1-sigmoid(alpha))*relu(...)
// ---------------------------------------------------------------------------
__global__ __launch_bounds__(128)
void sage_wmma_kernel(const float* __restrict__ X, const float* __restrict__ agg,
                      const float* __restrict__ cnt, int K,
                      const float* __restrict__ Ws, const float* __restrict__ bs,
                      const float* __restrict__ Wn, const float* __restrict__ bn,
                      float* __restrict__ H, const float* __restrict__ alphap,
                      int n_rows, int mode)
{
    const int NT = DOUT / 16;                                   // 16 col-tiles
    int wave  = blockIdx.x * (blockDim.x >> 5) + (threadIdx.x >> 5);
    int mtile = wave / NT;
    int ntile = wave - mtile * NT;
    int row0  = mtile * 16;
    if (row0 >= n_rows) return;                                 // wave-uniform
    int lane = threadIdx.x & 31;
    int half = lane >> 4;                                       // 0 or 1
    int m    = lane & 15;                                       // row (A) / col (B,C,D)
    int col0 = ntile * 16;

    const float* xr = X   + (long long)(row0 + m) * K + 2 * half;
    const float* ar = agg + (long long)(row0 + m) * K + 2 * half;
    float rc = 1.0f / (cnt[row0 + m] + 1e-12f);                 // scatter-mean scale

    v8f acc = {};
    // self term: X @ Ws
    const float* wsp = Ws + (long long)(2 * half) * DOUT + col0 + m;
    for (int k = 0; k < K; k += 4) {
        v2f a = *(const v2f*)(xr + k);                          // aligned float2
        v2f b;
        b.x = wsp[(long long)k * DOUT];
        b.y = wsp[(long long)(k + 1) * DOUT];
        acc = __builtin_amdgcn_wmma_f32_16x16x4_f32(false, a, false, b,
                                                    (short)0, acc, false, false);
    }
    // neighbor term: (agg * rcnt) @ Wn  (mean folded into A fragment)
    const float* wnp = Wn + (long long)(2 * half) * DOUT + col0 + m;
    for (int k = 0; k < K; k += 4) {
        v2f a = *(const v2f*)(ar + k);
        a.x *= rc; a.y *= rc;
        v2f b;
        b.x = wnp[(long long)k * DOUT];
        b.y = wnp[(long long)(k + 1) * DOUT];
        acc = __builtin_amdgcn_wmma_f32_16x16x4_f32(false, a, false, b,
                                                    (short)0, acc, false, false);
    }

    float bias = bs[col0 + m] + bn[col0 + m];
    float wb = 1.0f;
    if (mode) {
        float al = alphap[0];
        float sg = 1.0f / (1.0f + __expf(-al));
        wb = (mode == 1) ? sg : (1.0f - sg);
    }
    #pragma unroll
    for (int v = 0; v < 8; ++v) {
        int r = row0 + v + 8 * half;
        float val = acc[v] + bias;
        val = val > 0.0f ? val : 0.0f;
        float* o = H + (long long)r * DOUT + col0 + m;
        if (mode == 2)      *o += wb * val;
        else if (mode == 1) *o  = wb * val;
        else                *o  = val;
    }
}

// ---------------------------------------------------------------------------
// Host orchestration (graph-capture safe: only async memsets + launches)
// ---------------------------------------------------------------------------
extern "C" void kernel_launch(void* const* d_in, const int* in_sizes, int n_in,
                              void* d_out, int out_size, void* d_ws, size_t ws_size,
                              hipStream_t stream)
{
    const float* x     = (const float*)d_in[0];
    const float* alpha = (const float*)d_in[1];
    const float* s0_ws = (const float*)d_in[2];
    const float* s0_bs = (const float*)d_in[3];
    const float* s0_wn = (const float*)d_in[4];
    const float* s0_bn = (const float*)d_in[5];
    const float* s1_ws = (const float*)d_in[6];
    const float* s1_bs = (const float*)d_in[7];
    const float* s1_wn = (const float*)d_in[8];
    const float* s1_bn = (const float*)d_in[9];
    const float* a0_ws = (const float*)d_in[10];
    const float* a0_bs = (const float*)d_in[11];
    const float* a0_wn = (const float*)d_in[12];
    const float* a0_bn = (const float*)d_in[13];
    const float* a1_ws = (const float*)d_in[14];
    const float* a1_bs = (const float*)d_in[15];
    const float* a1_wn = (const float*)d_in[16];
    const float* a1_bn = (const float*)d_in[17];
    const int*   es    = (const int*)d_in[18];   // edge_spatial, shape (2,E)
    const int*   ea    = (const int*)d_in[19];   // edge_attr,    shape (2,E)

    const int DIN = 128;
    int N = in_sizes[0] / DIN;                   // 50000 (multiple of 16)
    int E = in_sizes[18] / 2;                    // 800000
    const int* srow = es;  const int* scol = es + E;
    const int* arow = ea;  const int* acol = ea + E;

    // scratch layout: cnt_s[N] | cnt_a[N] | agg[N*256] | h0[N*256]  (~103 MB)
    float* cnt_s = (float*)d_ws;
    float* cnt_a = cnt_s + N;
    float* agg   = cnt_a + N;
    float* h0    = agg + (long long)N * DOUT;
    float* out   = (float*)d_out;

    // per-stream edge counts (cnt_s and cnt_a are contiguous: one memset)
    hipMemsetAsync(cnt_s, 0, 2 * (size_t)N * sizeof(float), stream);
    int cb = (E + 255) / 256;
    count_edges_kernel<<<cb, 256, 0, stream>>>(srow, cnt_s, E);
    count_edges_kernel<<<cb, 256, 0, stream>>>(arow, cnt_a, E);

    int waves   = (N / 16) * (DOUT / 16);        // one wave per 16x16 tile
    int gblocks = (waves + 3) / 4;               // 128 threads = 4 waves (wave32)

    auto scat = [&](const float* src, const int* r, const int* c, int D) {
        hipMemsetAsync(agg, 0, (size_t)N * D * sizeof(float), stream);
        long long tt = (long long)E * (D >> 2);
        int sb = (int)((tt + 255) / 256);
        scatter_add_kernel<<<sb, 256, 0, stream>>>(src, r, c, agg, E, D);
    };

    // ---- spatial stream ----
    scat(x, srow, scol, DIN);
    sage_wmma_kernel<<<gblocks, 128, 0, stream>>>(x, agg, cnt_s, DIN,
        s0_ws, s0_bs, s0_wn, s0_bn, h0, alpha, N, 0);
    scat(h0, srow, scol, DOUT);
    sage_wmma_kernel<<<gblocks, 128, 0, stream>>>(h0, agg, cnt_s, DOUT,
        s1_ws, s1_bs, s1_wn, s1_bn, out, alpha, N, 1);   // out = w*relu(...)

    // ---- attribute stream ----
    scat(x, arow, acol, DIN);
    sage_wmma_kernel<<<gblocks, 128, 0, stream>>>(x, agg, cnt_a, DIN,
        a0_ws, a0_bs, a0_wn, a0_bn, h0, alpha, N, 0);
    scat(h0, arow, acol, DOUT);
    sage_wmma_kernel<<<gblocks, 128, 0, stream>>>(h0, agg, cnt_a, DOUT,
        a1_ws, a1_bs, a1_wn, a1_bn, out, alpha, N, 2);   // out += (1-w)*relu(...)
}